// GaussianWindow_39857296507051
// MI455X (gfx1250) — compile-verified
//
#include <hip/hip_runtime.h>

typedef __attribute__((ext_vector_type(2))) float v2f;
typedef __attribute__((ext_vector_type(8))) float v8f;

#define Bsz 8
#define Tsz 1024
#define Dsz 512
#define Ksz 10
#define Usz 1024
#define Vsz 80

// ---------------------------------------------------------------------------
// Kernel 1: abk = exp(x @ W^T + b); split into alpha/beta/kappa,
//           kappa += prev_kappa; one block (256 thr) per bt row.
// ---------------------------------------------------------------------------
__global__ void gw_abk_kernel(const float* __restrict__ x,
                              const float* __restrict__ W,
                              const float* __restrict__ bvec,
                              const float* __restrict__ prev_kappa,
                              float* __restrict__ alpha,
                              float* __restrict__ beta,
                              float* __restrict__ kappa,
                              float* __restrict__ kappa_out) {
    __shared__ float xs[Dsz];
    __shared__ float parts[30][8];
    const int bt = blockIdx.x;
    const int tid = threadIdx.x;

    const float* xrow = x + (size_t)bt * Dsz;
    xs[tid]       = xrow[tid];
    xs[tid + 256] = xrow[tid + 256];
    __syncthreads();

    if (tid < 240) {
        const int p = tid >> 3;     // output index 0..29
        const int c = tid & 7;      // chunk 0..7 (64 elems each)
        const float* wrow = W + (size_t)p * Dsz + c * 64;
        const float* xr = xs + c * 64;
        float s = 0.0f;
#pragma unroll
        for (int i = 0; i < 64; ++i) s = fmaf(xr[i], wrow[i], s);
        parts[p][c] = s;
    }
    __syncthreads();

    if (tid < 30) {
        float s = bvec[tid];
#pragma unroll
        for (int c = 0; c < 8; ++c) s += parts[tid][c];
        const float e = expf(s);
        if (tid < 10) {
            alpha[bt * Ksz + tid] = e;
        } else if (tid < 20) {
            beta[bt * Ksz + (tid - 10)] = e;
        } else {
            const int k = tid - 20;
            const float kp = e + prev_kappa[bt * Ksz + k];
            kappa[bt * Ksz + k] = kp;
            kappa_out[bt * Ksz + k] = kp;
        }
    }
}

// ---------------------------------------------------------------------------
// Kernel 2: phi[bt,u] = sum_k alpha*exp(-beta*(kappa-u)^2)
// grid = (B*T, U/256); block = 256
// ---------------------------------------------------------------------------
__global__ void gw_phi_kernel(const float* __restrict__ alpha,
                              const float* __restrict__ beta,
                              const float* __restrict__ kappa,
                              float* __restrict__ phi) {
    __shared__ float a_s[Ksz], b_s[Ksz], k_s[Ksz];
    const int bt = blockIdx.x;
    const int tid = threadIdx.x;
    if (tid < Ksz) {
        a_s[tid] = alpha[bt * Ksz + tid];
        b_s[tid] = beta[bt * Ksz + tid];
        k_s[tid] = kappa[bt * Ksz + tid];
    }
    __syncthreads();

    const int u = blockIdx.y * 256 + tid;
    const float uu = (float)u;
    float s = 0.0f;
#pragma unroll
    for (int k = 0; k < Ksz; ++k) {
        const float d = k_s[k] - uu;
        s += a_s[k] * expf(-b_s[k] * d * d);
    }
    phi[(size_t)bt * Usz + u] = s;
}

// ---------------------------------------------------------------------------
// Kernel 3: window[b] = phi[b] (1024x1024) @ onehot[b] (1024x80)
// fp32 WMMA 16x16x4. One wave per 16x16 output tile.
// Tiles: B * (T/16) * (V/16) = 8 * 64 * 5 = 2560 waves; 4 waves/block.
// ---------------------------------------------------------------------------
__global__ void gw_window_wmma_kernel(const float* __restrict__ phi,
                                      const float* __restrict__ onehot,
                                      float* __restrict__ win) {
    const int tid = threadIdx.x;
    const int wave_in_blk = tid >> 5;
    const int lane = tid & 31;
    const int half = lane >> 4;   // 0: lanes 0-15, 1: lanes 16-31
    const int lm = lane & 15;

    const int tile_id = blockIdx.x * 4 + wave_in_blk;   // 0..2559
    const int b  = tile_id / (64 * 5);
    const int r  = tile_id % (64 * 5);
    const int mt = r / 5;   // T tile
    const int nt = r % 5;   // V tile

    // A fragment source: phi row (mt*16 + lm), elements kk + 2*half + {0,1}
    const float* phiRow = phi + (size_t)b * Tsz * Usz + (size_t)(mt * 16 + lm) * Usz;
    // B fragment source: onehot column (nt*16 + lm), rows kk + 2*half + {0,1}
    const float* ohCol = onehot + (size_t)b * Usz * Vsz + (nt * 16 + lm);

    v8f acc = {};
    const int ko = 2 * half;
#pragma unroll 4
    for (int kk = 0; kk < Usz; kk += 4) {
        v2f a = *(const v2f*)(phiRow + kk + ko);          // contiguous pair, 8B aligned
        v2f bb;
        bb.x = ohCol[(size_t)(kk + ko) * Vsz];
        bb.y = ohCol[(size_t)(kk + ko + 1) * Vsz];
        acc = __builtin_amdgcn_wmma_f32_16x16x4_f32(
            /*neg_a=*/false, a, /*neg_b=*/false, bb,
            /*c_mod=*/(short)0, acc, /*reuse_a=*/false, /*reuse_b=*/false);
    }

    // D layout: VGPR v holds row (v + 8*half), col lm
    float* wout = win + (size_t)b * Tsz * Vsz
                      + (size_t)(mt * 16 + 8 * half) * Vsz
                      + (nt * 16 + lm);
#pragma unroll
    for (int v = 0; v < 8; ++v) wout[(size_t)v * Vsz] = acc[v];
}

// ---------------------------------------------------------------------------
extern "C" void kernel_launch(void* const* d_in, const int* in_sizes, int n_in,
                              void* d_out, int out_size, void* d_ws, size_t ws_size,
                              hipStream_t stream) {
    const float* x          = (const float*)d_in[0];  // (B,T,D)
    const float* onehot     = (const float*)d_in[1];  // (B,U,V)
    const float* prev_kappa = (const float*)d_in[2];  // (B,T,K,1)
    const float* W          = (const float*)d_in[3];  // (3K,D)
    const float* bvec       = (const float*)d_in[4];  // (3K,)

    float* out = (float*)d_out;
    float* win_out   = out;                                   // B*T*V
    float* kappa_out = out + (size_t)Bsz * Tsz * Vsz;         // B*T*K
    float* phi_out   = out + (size_t)Bsz * Tsz * Vsz
                           + (size_t)Bsz * Tsz * Ksz;         // B*T*U

    float* ws = (float*)d_ws;
    float* ws_alpha = ws;                                     // B*T*K
    float* ws_beta  = ws + (size_t)Bsz * Tsz * Ksz;
    float* ws_kappa = ws + (size_t)2 * Bsz * Tsz * Ksz;

    // K1: abk
    gw_abk_kernel<<<Bsz * Tsz, 256, 0, stream>>>(
        x, W, bvec, prev_kappa, ws_alpha, ws_beta, ws_kappa, kappa_out);

    // K2: phi
    gw_phi_kernel<<<dim3(Bsz * Tsz, Usz / 256), 256, 0, stream>>>(
        ws_alpha, ws_beta, ws_kappa, phi_out);

    // K3: window = phi @ onehot (fp32 WMMA); 2560 tiles, 4 waves/block
    gw_window_wmma_kernel<<<2560 / 4, 128, 0, stream>>>(phi_out, onehot, win_out);
}